// PosePost_44435731644718
// MI455X (gfx1250) — compile-verified
//
#include <hip/hip_runtime.h>

// ---------------------------------------------------------------------------
// PosePost: per-row (N=2M, 17 f32 in / 11 f32 out)
//   trans = p[0:3]; scale = exp(p[3:6]); weight = exp(p[16])
//   quat  = dominant eigenvector of 4x4 symmetric A(p[6:16])
// Memory-bound (224 MB @ 23.3 TB/s ~ 9.6 us). Uses CDNA5 async global<->LDS
// data movers (ASYNCcnt) for fully coalesced b128 traffic.
// ---------------------------------------------------------------------------

#define ROWS 128   // 2,000,000 % 128 == 0 -> no tail blocks for the real shape

#if defined(__gfx1250__) && \
    __has_builtin(__builtin_amdgcn_global_load_async_to_lds_b128) && \
    __has_builtin(__builtin_amdgcn_global_store_async_from_lds_b128)
#define HAVE_ASYNC_LDS 1
#else
#define HAVE_ASYNC_LDS 0
#endif

#if HAVE_ASYNC_LDS
// Builtin signature (from hipcc diagnostic): pointers to 16-byte int vectors.
typedef __attribute__((__vector_size__(16))) int v4i;
typedef __attribute__((address_space(1))) v4i GV4;   // global int4
typedef __attribute__((address_space(3))) v4i LV4;   // LDS int4

__device__ __forceinline__ void async_load16(const void* g, void* l) {
  // global_load_async_to_lds_b128  (tracked by ASYNCcnt)
  __builtin_amdgcn_global_load_async_to_lds_b128((GV4*)g, (LV4*)l, 0, 0);
}
__device__ __forceinline__ void async_store16(void* g, const void* l) {
  // global_store_async_from_lds_b128
  __builtin_amdgcn_global_store_async_from_lds_b128((GV4*)g, (LV4*)l, 0, 0);
}
__device__ __forceinline__ void wait_async0() {
#if __has_builtin(__builtin_amdgcn_s_wait_asynccnt)
  __builtin_amdgcn_s_wait_asynccnt(0);
#else
  asm volatile("s_wait_asynccnt 0" ::: "memory");
#endif
}
#endif // HAVE_ASYNC_LDS

__device__ __forceinline__ float frcp(float x) {
#if __has_builtin(__builtin_amdgcn_rcpf)
  return __builtin_amdgcn_rcpf(x);
#else
  return 1.0f / x;
#endif
}
__device__ __forceinline__ float frsq(float x) {
#if __has_builtin(__builtin_amdgcn_rsqf)
  return __builtin_amdgcn_rsqf(x);
#else
  return rsqrtf(x);
#endif
}

// Dominant eigenvector of 4x4 symmetric via shift + 5x matrix squaring.
// Symmetric storage m0..m9 = (00,01,02,03,11,12,13,22,23,33).
__device__ __forceinline__ void compute_row(const float* __restrict__ p,
                                            float o[11]) {
  o[0] = p[0]; o[1] = p[1]; o[2] = p[2];
  o[3] = __expf(p[3]); o[4] = __expf(p[4]); o[5] = __expf(p[5]);
  o[10] = __expf(p[16]);

  const float a = p[6], b = p[7], c = p[8], d = p[9], e = p[10];
  const float f = p[11], g = p[12], h = p[13], i2 = p[14], j = p[15];

  // Frobenius norm -> shift so B = A + s*I is PSD and lambda_max(A) dominates.
  float fro = a*a + e*e + h*h + j*j +
              2.0f * (b*b + c*c + d*d + f*f + g*g + i2*i2);
  float s = sqrtf(fro) + 1e-20f;

  float m0 = a + s, m1 = b, m2 = c, m3 = d;
  float m4 = e + s, m5 = f, m6 = g;
  float m7 = h + s, m8 = i2;
  float m9 = j + s;
  // trace(B) >= 2s > 0  -> safe normalization
  float iv = frcp(m0 + m4 + m7 + m9);
  m0 *= iv; m1 *= iv; m2 *= iv; m3 *= iv; m4 *= iv;
  m5 *= iv; m6 *= iv; m7 *= iv; m8 *= iv; m9 *= iv;

  // B^(2^5) = B^32: symmetric square, 10 dot-4s per step, renorm by trace.
#pragma unroll
  for (int it = 0; it < 5; ++it) {
    float t0 = m0*m0 + m1*m1 + m2*m2 + m3*m3;
    float t1 = m0*m1 + m1*m4 + m2*m5 + m3*m6;
    float t2 = m0*m2 + m1*m5 + m2*m7 + m3*m8;
    float t3 = m0*m3 + m1*m6 + m2*m8 + m3*m9;
    float t4 = m1*m1 + m4*m4 + m5*m5 + m6*m6;
    float t5 = m1*m2 + m4*m5 + m5*m7 + m6*m8;
    float t6 = m1*m3 + m4*m6 + m5*m8 + m6*m9;
    float t7 = m2*m2 + m5*m5 + m7*m7 + m8*m8;
    float t8 = m2*m3 + m5*m6 + m7*m8 + m8*m9;
    float t9 = m3*m3 + m6*m6 + m8*m8 + m9*m9;
    float r = frcp(t0 + t4 + t7 + t9 + 1e-32f);
    m0 = t0*r; m1 = t1*r; m2 = t2*r; m3 = t3*r; m4 = t4*r;
    m5 = t5*r; m6 = t6*r; m7 = t7*r; m8 = t8*r; m9 = t9*r;
  }

  // M ~ lambda^32 * v v^T: take column with largest diagonal, normalize.
  float v0 = m0, v1 = m1, v2 = m2, v3 = m3, best = m0;
  if (m4 > best) { best = m4; v0 = m1; v1 = m4; v2 = m5; v3 = m6; }
  if (m7 > best) { best = m7; v0 = m2; v1 = m5; v2 = m7; v3 = m8; }
  if (m9 > best) { best = m9; v0 = m3; v1 = m6; v2 = m8; v3 = m9; }
  float n = frsq(v0*v0 + v1*v1 + v2*v2 + v3*v3 + 1e-38f);
  o[6] = v0 * n; o[7] = v1 * n; o[8] = v2 * n; o[9] = v3 * n;
}

__global__ void __launch_bounds__(ROWS)
pose_post_kernel(const float* __restrict__ ep, float* __restrict__ out, int N) {
  const int tid = threadIdx.x;
  const int rowBase = (int)blockIdx.x * ROWS;

#if HAVE_ASYNC_LDS
  // SoA output staging bases (dwords) inside ldsOut
  constexpr int SB = 3 * ROWS;    // scale
  constexpr int QB = 6 * ROWS;    // quat
  constexpr int WB = 10 * ROWS;   // weight
  __shared__ __align__(16) float ldsIn[ROWS * 17];   // 8704 B, 16B-aligned tile
  __shared__ __align__(16) float ldsOut[ROWS * 11];  // 5632 B

  if (rowBase + ROWS <= N && (N & 3) == 0) {
    // ---- async coalesced tile load: 544 x b128 chunks, contiguous, aligned
    const float* gsrc = ep + (size_t)rowBase * 17;
#pragma unroll
    for (int i = 0; i < 5; ++i) {
      int idx = tid + i * ROWS;
      if (idx < (ROWS * 17) / 4)
        async_load16(gsrc + idx * 4, &ldsIn[idx * 4]);
    }
    wait_async0();
    __syncthreads();

    // ---- compute (LDS row reads stride 17 dwords: gcd(17,64)=1, no conflicts)
    float o[11];
    compute_row(&ldsIn[tid * 17], o);

    ldsOut[3 * tid + 0] = o[0];
    ldsOut[3 * tid + 1] = o[1];
    ldsOut[3 * tid + 2] = o[2];
    ldsOut[SB + 3 * tid + 0] = o[3];
    ldsOut[SB + 3 * tid + 1] = o[4];
    ldsOut[SB + 3 * tid + 2] = o[5];
    ldsOut[QB + 4 * tid + 0] = o[6];
    ldsOut[QB + 4 * tid + 1] = o[7];
    ldsOut[QB + 4 * tid + 2] = o[8];
    ldsOut[QB + 4 * tid + 3] = o[9];
    ldsOut[WB + tid] = o[10];
    __syncthreads();

    // ---- async coalesced stores: 4 contiguous 16B-aligned regions per block
    float* gT = out + (size_t)3 * rowBase;
    float* gS = out + (size_t)3 * N + (size_t)3 * rowBase;
    float* gQ = out + (size_t)6 * N + (size_t)4 * rowBase;
    float* gW = out + (size_t)10 * N + (size_t)rowBase;
    if (tid < (3 * ROWS) / 4) {                      // 96 chunks each
      async_store16(gT + tid * 4, &ldsOut[tid * 4]);
      async_store16(gS + tid * 4, &ldsOut[SB + tid * 4]);
    }
    async_store16(gQ + tid * 4, &ldsOut[QB + tid * 4]);   // 128 chunks
    if (tid < ROWS / 4)                                   // 32 chunks
      async_store16(gW + tid * 4, &ldsOut[WB + tid * 4]);
    wait_async0();
    return;
  }
#endif // HAVE_ASYNC_LDS

  // ---- generic tail / fallback path (uniform branch per block)
  int r = rowBase + tid;
  if (r >= N) return;
  float p[17];
  const float* src = ep + (size_t)r * 17;
#pragma unroll
  for (int k = 0; k < 17; ++k) p[k] = src[k];
  float o[11];
  compute_row(p, o);
  out[(size_t)3 * r + 0] = o[0];
  out[(size_t)3 * r + 1] = o[1];
  out[(size_t)3 * r + 2] = o[2];
  out[(size_t)3 * N + 3 * r + 0] = o[3];
  out[(size_t)3 * N + 3 * r + 1] = o[4];
  out[(size_t)3 * N + 3 * r + 2] = o[5];
  out[(size_t)6 * N + 4 * r + 0] = o[6];
  out[(size_t)6 * N + 4 * r + 1] = o[7];
  out[(size_t)6 * N + 4 * r + 2] = o[8];
  out[(size_t)6 * N + 4 * r + 3] = o[9];
  out[(size_t)10 * N + r] = o[10];
}

extern "C" void kernel_launch(void* const* d_in, const int* in_sizes, int n_in,
                              void* d_out, int out_size, void* d_ws,
                              size_t ws_size, hipStream_t stream) {
  (void)n_in; (void)out_size; (void)d_ws; (void)ws_size;
  const float* ep = (const float*)d_in[0];
  float* out = (float*)d_out;
  const int N = in_sizes[0] / 17;   // 2,000,000
  const int blocks = (N + ROWS - 1) / ROWS;
  pose_post_kernel<<<blocks, ROWS, 0, stream>>>(ep, out, N);
}